// FeatureWeightNet_76106820485560
// MI455X (gfx1250) — compile-verified
//
#include <hip/hip_runtime.h>
#include <math.h>

typedef float v2f __attribute__((ext_vector_type(2)));
typedef float v8f __attribute__((ext_vector_type(8)));

#define NB   9
#define EPSB 1e-5f
#define B_   4
#define C_   64
#define H_   128
#define W_   160
#define HW_  (H_ * W_)
// points = B*N*H*W = 737280; 128 points per block (8 waves x 16 points)
#define NPOINTS (B_ * NB * H_ * W_)
#define PT_PER_BLOCK 128
#define LDS_STRIDE 17

// Wave-local LDS fence: each wave owns a private LDS tile, and DS ops from a
// single wave execute in order, so no workgroup barrier is needed between the
// transpose store/load phases. The "memory" clobber stops the compiler from
// hoisting the ds_loads above the ds_stores; s_wait_dscnt 0 guarantees the
// stores have drained.
#define WAVE_LDS_FENCE() asm volatile("s_wait_dscnt 0x0" ::: "memory")

__global__ __launch_bounds__(256) void fwnet_fused_kernel(
    const float* __restrict__ feat, const float* __restrict__ grid,
    const float* __restrict__ w0,  const float* __restrict__ g0,
    const float* __restrict__ b0,  const float* __restrict__ m0,
    const float* __restrict__ v0,  const float* __restrict__ w1,
    const float* __restrict__ g1,  const float* __restrict__ b1,
    const float* __restrict__ m1,  const float* __restrict__ v1,
    const float* __restrict__ wsim, const float* __restrict__ bsim,
    float* __restrict__ out)
{
    __shared__ float lds[8 * 16 * LDS_STRIDE];
    const int lane = threadIdx.x & 31;
    const int wave = threadIdx.x >> 5;
    const int n  = lane & 15;   // column (out-channel / point row) owned by lane
    const int hh = lane >> 4;   // lane half: channel split + WMMA K/M half
    float* wl = lds + wave * 16 * LDS_STRIDE;

    // ---------------- per-lane constant operands ----------------
    // Layer-0 B operand: w0 is [16 out, 8 in]; B_j(k, col=n) = w0[n][4j+k].
    // B layout (32-bit, 4x16): VGPR v, lanes0-15 hold K=v, lanes16-31 K=v+2.
    v2f b00, b01;
    b00.x = w0[n * 8 + 0 + 2 * hh];
    b00.y = w0[n * 8 + 1 + 2 * hh];
    b01.x = w0[n * 8 + 4 + 2 * hh];
    b01.y = w0[n * 8 + 5 + 2 * hh];
    // Layer-1 B operands: w1 is [8 out, 16 in], zero-padded to 16 outs.
    v2f b1j[4];
#pragma unroll
    for (int j = 0; j < 4; ++j) {
        b1j[j].x = (n < 8) ? w1[n * 16 + 4 * j + 2 * hh + 0] : 0.0f;
        b1j[j].y = (n < 8) ? w1[n * 16 + 4 * j + 2 * hh + 1] : 0.0f;
    }
    // Folded BN params (per out-channel n)
    const float inv0 = g0[n] * rsqrtf(v0[n] + EPSB);
    const float sh0  = b0[n] - m0[n] * inv0;
    const int n8 = (n < 8) ? n : 0;
    const float inv1 = g1[n8] * rsqrtf(v1[n8] + EPSB);
    const float sh1  = b1[n8] - m1[n8] * inv1;

    // ---------------- gather + per-group correlation ----------------
    const int p  = blockIdx.x * PT_PER_BLOCK + wave * 16 + n;   // point id
    const int hw = p % HW_;                                     // (h,w) in-plane
    const int b  = p / (NB * HW_);

    const float gx = grid[2 * p + 0];
    const float gy = grid[2 * p + 1];
    float ix = fminf(fmaxf(((gx + 1.0f) * (float)W_ - 1.0f) * 0.5f, 0.0f), (float)(W_ - 1));
    float iy = fminf(fmaxf(((gy + 1.0f) * (float)H_ - 1.0f) * 0.5f, 0.0f), (float)(H_ - 1));
    const float x0f = floorf(ix), y0f = floorf(iy);
    const float fx = ix - x0f, fy = iy - y0f;
    const int x0i = (int)x0f, y0i = (int)y0f;
    const int x1i = min(x0i + 1, W_ - 1), y1i = min(y0i + 1, H_ - 1);
    const float wgt11 = fx * fy;
    const float wgt10 = fy - wgt11;               // (1-fx)*fy
    const float wgt01 = fx - wgt11;               // fx*(1-fy)
    const float wgt00 = 1.0f - fx - fy + wgt11;   // (1-fx)*(1-fy)
    const int o00 = y0i * W_ + x0i;
    const int o01 = y0i * W_ + x1i;
    const int o10 = y1i * W_ + x0i;
    const int o11 = y1i * W_ + x1i;

    // this lane handles channels [hh*32, hh*32+32) -> groups [hh*4, hh*4+4)
    const float* fb = feat + ((size_t)b * C_ + (size_t)hh * 32) * HW_;
    float acc[4] = {0.f, 0.f, 0.f, 0.f};
#pragma unroll 16
    for (int c = 0; c < 32; ++c) {
        const float* fc = fb + (size_t)c * HW_;
        float s = fmaf(wgt00, fc[o00],
                  fmaf(wgt01, fc[o01],
                  fmaf(wgt10, fc[o10], wgt11 * fc[o11])));
        acc[c >> 3] = fmaf(s, fc[hw], acc[c >> 3]);
    }
#pragma unroll
    for (int q = 0; q < 4; ++q) acc[q] *= 0.125f;   // mean over C/G = 8

    // Redistribute group sums into 16x4 fp32 A layout:
    // A(m,k): VGPR k&1, lanes0-15 hold K=0,1; lanes16-31 hold K=2,3.
    const float sw0 = __shfl_xor(acc[0], 16, 32);
    const float sw1 = __shfl_xor(acc[1], 16, 32);
    const float sw2 = __shfl_xor(acc[2], 16, 32);
    const float sw3 = __shfl_xor(acc[3], 16, 32);
    v2f a0, a1;
    a0.x = hh ? sw2    : acc[0];   // K=0..3  (groups 0..3)
    a0.y = hh ? sw3    : acc[1];
    a1.x = hh ? acc[2] : sw0;      // K=4..7  (groups 4..7)
    a1.y = hh ? acc[3] : sw1;

    // Layer 0: [16 pts x 8] @ w0^T[8 x 16] via two fp32 16x16x4 WMMAs
    v8f c0 = {};
    c0 = __builtin_amdgcn_wmma_f32_16x16x4_f32(false, a0, false, b00, (short)0, c0, false, false);
    c0 = __builtin_amdgcn_wmma_f32_16x16x4_f32(false, a1, false, b01, (short)0, c0, false, false);

    // BN0 + ReLU (channel = n for every row element)
#pragma unroll
    for (int r = 0; r < 8; ++r)
        c0[r] = fmaxf(fmaf(c0[r], inv0, sh0), 0.0f);

    // D-layout -> A-layout transpose through this wave's private LDS tile
    // (row m = r + 8*hh, col n)
#pragma unroll
    for (int r = 0; r < 8; ++r)
        wl[(r + 8 * hh) * LDS_STRIDE + n] = c0[r];
    WAVE_LDS_FENCE();

    // Layer 1: [16 pts x 16] @ w1^T(padded)[16 x 16] via four WMMAs
    v8f c1 = {};
#pragma unroll
    for (int j = 0; j < 4; ++j) {
        v2f a;
        a.x = wl[n * LDS_STRIDE + 4 * j + 2 * hh + 0];
        a.y = wl[n * LDS_STRIDE + 4 * j + 2 * hh + 1];
        c1 = __builtin_amdgcn_wmma_f32_16x16x4_f32(false, a, false, b1j[j], (short)0, c1, false, false);
    }
    WAVE_LDS_FENCE();   // loads done before the tile is overwritten below

    // BN1 + ReLU, then stash for the final 8-dot
#pragma unroll
    for (int r = 0; r < 8; ++r)
        c1[r] = fmaxf(fmaf(c1[r], inv1, sh1), 0.0f);
#pragma unroll
    for (int r = 0; r < 8; ++r)
        wl[(r + 8 * hh) * LDS_STRIDE + n] = c1[r];
    WAVE_LDS_FENCE();

    // Final: sigmoid(w_sim . x + b_sim); lanes 0-15 each own one point ->
    // 16 contiguous coalesced stores per wave.
    if (hh == 0) {
        float s = bsim[0];
#pragma unroll
        for (int c = 0; c < 8; ++c)
            s = fmaf(wl[n * LDS_STRIDE + c], wsim[c], s);
        out[p] = 1.0f / (1.0f + __expf(-s));
    }
}

extern "C" void kernel_launch(void* const* d_in, const int* in_sizes, int n_in,
                              void* d_out, int out_size, void* d_ws, size_t ws_size,
                              hipStream_t stream) {
    (void)in_sizes; (void)n_in; (void)out_size; (void)d_ws; (void)ws_size;
    const float* feat = (const float*)d_in[0];
    const float* grid = (const float*)d_in[1];
    const float* w0   = (const float*)d_in[2];
    const float* g0   = (const float*)d_in[3];
    const float* b0   = (const float*)d_in[4];
    const float* m0   = (const float*)d_in[5];
    const float* v0   = (const float*)d_in[6];
    const float* w1   = (const float*)d_in[7];
    const float* g1   = (const float*)d_in[8];
    const float* b1   = (const float*)d_in[9];
    const float* m1   = (const float*)d_in[10];
    const float* v1   = (const float*)d_in[11];
    const float* wsim = (const float*)d_in[12];
    const float* bsim = (const float*)d_in[13];
    float* out = (float*)d_out;

    const int nblocks = NPOINTS / PT_PER_BLOCK;   // 737280 / 128 = 5760
    fwnet_fused_kernel<<<nblocks, 256, 0, stream>>>(
        feat, grid, w0, g0, b0, m0, v0, w1, g1, b1, m1, v1, wsim, bsim, out);
}